// ResidualBlock_simple_61229053772003
// MI455X (gfx1250) — compile-verified
//
#include <hip/hip_runtime.h>
#include <math.h>

// ------------------------------------------------------------------
// Problem constants (from reference)
// ------------------------------------------------------------------
#define NB    2
#define LSEQ  1024
#define DM    768          // d_model
#define DI    1536         // d_inner
#define NST   16           // n_state == HID
#define DTR   48           // dt_rank
#define MROWS (NB * LSEQ)  // 2048 token rows

typedef float v2f __attribute__((ext_vector_type(2)));
typedef float v8f __attribute__((ext_vector_type(8)));

// ------------------------------------------------------------------
// WMMA f32 16x16x4 core: one wave computes a 16x16 f32 output tile of
//   D[row0:row0+16, col0:col0+16] = A[row0: , :K] * W[col0: , :K]^T
// A is (M x K) row-major, W is (N x K) row-major (i.e. weight[j][k]).
// Lane layout per CDNA5 ISA 7.12.2:
//   A 16x4 tile : lanes 0-15 & 16-31 both hold rows M=0..15 (row = lane&15),
//                 VGPR0/1 hold K = {0,1} (lanes 0-15) or {2,3} (lanes 16-31).
//   B 4x16 tile : mirror — lane holds column N = lane&15, K split the same way.
// ------------------------------------------------------------------
__device__ __forceinline__ v8f wmma_tile_f32(const float* __restrict__ A, int lda,
                                             const float* __restrict__ W, int ldw,
                                             int K, int row0, int col0) {
  const int lane = threadIdx.x & 31;
  const int half = lane >> 4;   // 0 => K 0,1   1 => K 2,3
  const int l16  = lane & 15;
  const float* ap = A + (size_t)(row0 + l16) * lda + 2 * half;
  const float* wp = W + (size_t)(col0 + l16) * ldw + 2 * half;
  v8f acc = (v8f){0.f, 0.f, 0.f, 0.f, 0.f, 0.f, 0.f, 0.f};
  for (int k = 0; k < K; k += 4) {
    v2f a = *(const v2f*)(ap + k);   // 8B-aligned: lda even, k%4==0, 2*half even
    v2f b = *(const v2f*)(wp + k);
    acc = __builtin_amdgcn_wmma_f32_16x16x4_f32(
        /*neg_a=*/false, a, /*neg_b=*/false, b,
        /*c_mod=*/(short)0, acc, /*reuse_a=*/false, /*reuse_b=*/false);
  }
  return acc;
}

// C/D layout: VGPR r -> row = r + 8*(lane>=16), col = lane&15
__device__ __forceinline__ int wmma_out_row(int r) {
  return r + 8 * ((threadIdx.x & 31) >> 4);
}
__device__ __forceinline__ int wmma_out_col() { return threadIdx.x & 15; }

__device__ __forceinline__ float silu_f(float v) {
  return v / (1.f + __expf(-v));
}

// ------------------------------------------------------------------
// 1) RMSNorm:  xn = x * rsqrt(mean(x^2)+1e-5) * rms_w     (one block / row)
// ------------------------------------------------------------------
__global__ void rb_rms_kernel(const float* __restrict__ x,
                              const float* __restrict__ w,
                              float* __restrict__ xn) {
  __shared__ float red[256];
  const int m = blockIdx.x;
  const float* xr = x + (size_t)m * DM;
  float s = 0.f;
  for (int c = threadIdx.x; c < DM; c += 256) { float v = xr[c]; s += v * v; }
  red[threadIdx.x] = s;
  __syncthreads();
  for (int st = 128; st > 0; st >>= 1) {
    if (threadIdx.x < st) red[threadIdx.x] += red[threadIdx.x + st];
    __syncthreads();
  }
  const float r = rsqrtf(red[0] * (1.f / DM) + 1e-5f);
  for (int c = threadIdx.x; c < DM; c += 256)
    xn[(size_t)m * DM + c] = xr[c] * r * w[c];
}

// ------------------------------------------------------------------
// 2) GEMM1:  xr(2048x3072) = xn(2048x768) @ Win(3072x768)^T    [WMMA]
// ------------------------------------------------------------------
__global__ void rb_gemm_xr_kernel(const float* __restrict__ xn,
                                  const float* __restrict__ Win,
                                  float* __restrict__ xr) {
  const int NT = (2 * DI) / 16;                       // 192 col tiles
  const int tile = blockIdx.x * 8 + (threadIdx.x >> 5);
  const int tx = tile % NT, ty = tile / NT;
  v8f acc = wmma_tile_f32(xn, DM, Win, DM, DM, ty * 16, tx * 16);
  const int col = tx * 16 + wmma_out_col();
#pragma unroll
  for (int r = 0; r < 8; ++r)
    xr[(size_t)(ty * 16 + wmma_out_row(r)) * (2 * DI) + col] = acc[r];
}

// ------------------------------------------------------------------
// 3) Depthwise causal conv(4) + bias + SiLU  (u = xr[:, :DI])
//    Six blocks == one token row (256 threads x 6 blocks span DI)
// ------------------------------------------------------------------
__global__ void rb_conv_silu_kernel(const float* __restrict__ xr,
                                    const float* __restrict__ cw,
                                    const float* __restrict__ cb,
                                    float* __restrict__ uc) {
  const size_t idx = (size_t)blockIdx.x * 256 + threadIdx.x;  // over MROWS*DI
  const int c = (int)(idx % DI);
  const int m = (int)(idx / DI);
  const int l = m & (LSEQ - 1);
  const size_t base = (size_t)m * (2 * DI) + c;
  const float w0 = cw[c * 4 + 0], w1 = cw[c * 4 + 1];
  const float w2 = cw[c * 4 + 2], w3 = cw[c * 4 + 3];
  float acc = cb[c] + xr[base] * w3;
  if (l >= 1) acc += xr[base - (size_t)(2 * DI)] * w2;
  if (l >= 2) acc += xr[base - (size_t)2 * (2 * DI)] * w1;
  if (l >= 3) acc += xr[base - (size_t)3 * (2 * DI)] * w0;
  uc[(size_t)m * DI + c] = silu_f(acc);
}

// ------------------------------------------------------------------
// 4) GEMM2: x_dbl(2048x80) = uc(2048x1536) @ Wx(80x1536)^T     [WMMA]
// ------------------------------------------------------------------
__global__ void rb_gemm_xdbl_kernel(const float* __restrict__ uc,
                                    const float* __restrict__ Wx,
                                    float* __restrict__ xdbl) {
  const int NT = 80 / 16;                             // 5 col tiles
  const int tile = blockIdx.x * 8 + (threadIdx.x >> 5);
  const int tx = tile % NT, ty = tile / NT;
  v8f acc = wmma_tile_f32(uc, DI, Wx, DI, DI, ty * 16, tx * 16);
  const int col = tx * 16 + wmma_out_col();
#pragma unroll
  for (int r = 0; r < 8; ++r)
    xdbl[(size_t)(ty * 16 + wmma_out_row(r)) * 80 + col] = acc[r];
}

// ------------------------------------------------------------------
// 5) GEMM3: delta = softplus(x_dbl[:, :48] @ Wdt(1536x48)^T + bdt)   [WMMA]
// ------------------------------------------------------------------
__global__ void rb_gemm_delta_kernel(const float* __restrict__ xdbl,
                                     const float* __restrict__ Wdt,
                                     const float* __restrict__ bdt,
                                     float* __restrict__ delta) {
  const int NT = DI / 16;                             // 96 col tiles
  const int tile = blockIdx.x * 8 + (threadIdx.x >> 5);
  const int tx = tile % NT, ty = tile / NT;
  v8f acc = wmma_tile_f32(xdbl, 80, Wdt, DTR, DTR, ty * 16, tx * 16);
  const int col = tx * 16 + wmma_out_col();
  const float bias = bdt[col];
#pragma unroll
  for (int r = 0; r < 8; ++r) {
    float v = acc[r] + bias;
    float sp = (v > 15.f) ? v : log1pf(__expf(v));    // softplus
    delta[(size_t)(ty * 16 + wmma_out_row(r)) * DI + col] = sp;
  }
}

// ------------------------------------------------------------------
// 6) Fold LN affine into Wq/Wk/Wv:
//    W'[j][n] = W[j][n]*g[n];  rowsum[j] = sum_n W'[j][n];  bdot[j]=sum_n W[j][n]*b[n]
//    fold layout (floats): [0:256) Wq' [256:512) Wk' [512:768) Wv'
//                          [768:816) rowsums q,k,v   [816:864) bdots q,k,v
// ------------------------------------------------------------------
__global__ void rb_prep_fold_kernel(const float* __restrict__ Wq,
                                    const float* __restrict__ Wk,
                                    const float* __restrict__ Wv,
                                    const float* __restrict__ gs,
                                    const float* __restrict__ bs,
                                    const float* __restrict__ gc,
                                    const float* __restrict__ bc,
                                    float* __restrict__ fold) {
  const int tid = threadIdx.x;
  if (tid >= 48) return;
  const int mat = tid >> 4, j = tid & 15;
  const float* W = (mat == 0) ? Wq : ((mat == 1) ? Wk : Wv);
  const float* g = (mat == 0) ? gs : gc;
  const float* b = (mat == 0) ? bs : bc;
  float rs = 0.f, bd = 0.f;
  for (int n = 0; n < 16; ++n) {
    const float wf = W[j * 16 + n] * g[n];
    fold[mat * 256 + j * 16 + n] = wf;
    rs += wf;
    bd += W[j * 16 + n] * b[n];
  }
  fold[768 + mat * 16 + j] = rs;
  fold[816 + mat * 16 + j] = bd;
}

// ------------------------------------------------------------------
// 7) Sequential SSM scan, one lane per (b,d) channel (3072 lanes).
//    dA/dBu computed in-register (saves 2x192MB of HBM traffic).
//    Output fused:  yf = (y_scan + uc*Dp) * silu(res)
// ------------------------------------------------------------------
__global__ void rb_scan_kernel(const float* __restrict__ delta,
                               const float* __restrict__ uc,
                               const float* __restrict__ xr,
                               const float* __restrict__ xdbl,
                               const float* __restrict__ A_log,
                               const float* __restrict__ Dp,
                               const float* __restrict__ fold,
                               float* __restrict__ yf) {
  __shared__ float sw[864];
  for (int i = threadIdx.x; i < 864; i += 256) sw[i] = fold[i];
  __syncthreads();

  const int b = blockIdx.x / 6;                         // 6 blocks per batch
  const int d = (blockIdx.x % 6) * 256 + threadIdx.x;   // channel
  const float dscale = Dp[d];

  float Arow[16];
#pragma unroll
  for (int n = 0; n < 16; ++n) Arow[n] = -__expf(A_log[(size_t)d * 16 + n]);

  float st[16];
#pragma unroll
  for (int n = 0; n < 16; ++n) st[n] = 0.f;

  const size_t mb = (size_t)b * LSEQ;
  for (int t = 0; t < LSEQ; ++t) {
    const size_t m = mb + t;
    const float dt = delta[m * DI + d];
    const float ut = uc[m * DI + d];
    const float rt = xr[m * (2 * DI) + DI + d];         // res stream
    if (t + 1 < LSEQ) {                                 // global_prefetch_b8
      __builtin_prefetch(&delta[(m + 1) * DI + d], 0, 1);
      __builtin_prefetch(&uc[(m + 1) * DI + d], 0, 1);
    }
    float Bv[16], Cv[16];
    const float4* bc4 = (const float4*)(xdbl + m * 80 + DTR);
    *(float4*)&Bv[0] = bc4[0];  *(float4*)&Bv[4]  = bc4[1];
    *(float4*)&Bv[8] = bc4[2];  *(float4*)&Bv[12] = bc4[3];
    *(float4*)&Cv[0] = bc4[4];  *(float4*)&Cv[4]  = bc4[5];
    *(float4*)&Cv[8] = bc4[6];  *(float4*)&Cv[12] = bc4[7];

    float ctrl[16];
    float ss = 0.f, s2 = 0.f, cs = 0.f, c2 = 0.f;
#pragma unroll
    for (int n = 0; n < 16; ++n) {
      st[n] *= __expf(dt * Arow[n]);                    // state = dA * st
      ss += st[n];  s2 += st[n] * st[n];
      ctrl[n] = dt * Bv[n] * ut;                        // control = dBu
      cs += ctrl[n]; c2 += ctrl[n] * ctrl[n];
    }
    const float msn = ss * 0.0625f;
    const float rs  = rsqrtf(fmaxf(s2 * 0.0625f - msn * msn, 0.f) + 1e-5f);
    const float mcn = cs * 0.0625f;
    const float rc  = rsqrtf(fmaxf(c2 * 0.0625f - mcn * mcn, 0.f) + 1e-5f);

    float qv[16], vv[16];
    float wdot = 0.f;
#pragma unroll
    for (int j = 0; j < 16; ++j) {
      const float* wq = &sw[j * 16];                    // uniform-addr LDS bcast
      const float* wk = &sw[256 + j * 16];
      const float* wv = &sw[512 + j * 16];
      float dq = 0.f, dk = 0.f, dv = 0.f;
#pragma unroll
      for (int n = 0; n < 16; ++n) {
        dq += st[n]   * wq[n];
        dk += ctrl[n] * wk[n];
        dv += ctrl[n] * wv[n];
      }
      const float q = rs * (dq - msn * sw[768 + j]) + sw[816 + j];
      const float k = rc * (dk - mcn * sw[784 + j]) + sw[832 + j];
      const float v = rc * (dv - mcn * sw[800 + j]) + sw[848 + j];
      qv[j] = q; vv[j] = v; wdot += q * k;
    }
    const float w = wdot * 0.25f;                       // 1/sqrt(HID)
    float y = 0.f;
#pragma unroll
    for (int j = 0; j < 16; ++j) {
      const float nj = qv[j] + w * (vv[j] - qv[j]);     // (1-w)q + w v
      st[j] = nj;
      y += nj * Cv[j];
    }
    yf[m * DI + d] = (y + ut * dscale) * silu_f(rt);
  }
}

// ------------------------------------------------------------------
// 8) GEMM4: h(2048x768) = yf(2048x1536) @ Wout(768x1536)^T      [WMMA]
// ------------------------------------------------------------------
__global__ void rb_gemm_h_kernel(const float* __restrict__ yf,
                                 const float* __restrict__ Wout,
                                 float* __restrict__ h) {
  const int NT = DM / 16;                               // 48
  const int tile = blockIdx.x * 8 + (threadIdx.x >> 5);
  const int tx = tile % NT, ty = tile / NT;
  v8f acc = wmma_tile_f32(yf, DI, Wout, DI, DI, ty * 16, tx * 16);
  const int col = tx * 16 + wmma_out_col();
#pragma unroll
  for (int r = 0; r < 8; ++r)
    h[(size_t)(ty * 16 + wmma_out_row(r)) * DM + col] = acc[r];
}

// ------------------------------------------------------------------
// 9) emb path: eo(2x1536) = silu(emb) @ We(1536x1024)^T + be
// ------------------------------------------------------------------
__global__ void rb_emb_kernel(const float* __restrict__ emb,
                              const float* __restrict__ We,
                              const float* __restrict__ be,
                              float* __restrict__ eo) {
  const int j = blockIdx.x * 256 + threadIdx.x;
  if (j >= NB * 2 * DM) return;
  const int b = j / (2 * DM), c = j % (2 * DM);
  const float* er = emb + (size_t)b * 1024;
  const float* wr = We + (size_t)c * 1024;
  float acc = be[c];
  for (int k = 0; k < 1024; ++k) acc += silu_f(er[k]) * wr[k];
  eo[j] = acc;
}

// ------------------------------------------------------------------
// 10) h2 = LN(h)*(1+scale)+shift ; g = silu(h2)      (one block / row)
// ------------------------------------------------------------------
__global__ void rb_lnfinal_kernel(const float* __restrict__ h,
                                  const float* __restrict__ gamma,
                                  const float* __restrict__ beta,
                                  const float* __restrict__ eo,
                                  float* __restrict__ outg) {
  __shared__ float red[256];
  __shared__ float red2[256];
  const int m = blockIdx.x;
  const int b = m >> 10;
  const float* hr = h + (size_t)m * DM;
  float s = 0.f, s2 = 0.f;
  for (int c = threadIdx.x; c < DM; c += 256) { float v = hr[c]; s += v; s2 += v * v; }
  red[threadIdx.x] = s; red2[threadIdx.x] = s2;
  __syncthreads();
  for (int st = 128; st > 0; st >>= 1) {
    if (threadIdx.x < st) {
      red[threadIdx.x]  += red[threadIdx.x + st];
      red2[threadIdx.x] += red2[threadIdx.x + st];
    }
    __syncthreads();
  }
  const float mean = red[0] * (1.f / DM);
  const float var  = red2[0] * (1.f / DM) - mean * mean;
  const float r = rsqrtf(var + 1e-5f);
  const float* scale = eo + (size_t)b * (2 * DM);
  const float* shift = scale + DM;
  for (int c = threadIdx.x; c < DM; c += 256) {
    float z = (hr[c] - mean) * r * gamma[c] + beta[c];
    z = z * (1.f + scale[c]) + shift[c];
    outg[(size_t)m * DM + c] = silu_f(z);
  }
}

// ------------------------------------------------------------------
// 11) GEMM5 + residual: out = x + g @ Wo(768x768)^T + bo        [WMMA]
// ------------------------------------------------------------------
__global__ void rb_gemm_out_kernel(const float* __restrict__ g,
                                   const float* __restrict__ Wo,
                                   const float* __restrict__ x,
                                   const float* __restrict__ bo,
                                   float* __restrict__ out) {
  const int NT = DM / 16;                               // 48
  const int tile = blockIdx.x * 8 + (threadIdx.x >> 5);
  const int tx = tile % NT, ty = tile / NT;
  v8f acc = wmma_tile_f32(g, DM, Wo, DM, DM, ty * 16, tx * 16);
  const int col = tx * 16 + wmma_out_col();
  const float bias = bo[col];
#pragma unroll
  for (int r = 0; r < 8; ++r) {
    const size_t idx = (size_t)(ty * 16 + wmma_out_row(r)) * DM + col;
    out[idx] = x[idx] + acc[r] + bias;
  }
}

// ------------------------------------------------------------------
// Launch
// ------------------------------------------------------------------
extern "C" void kernel_launch(void* const* d_in, const int* in_sizes, int n_in,
                              void* d_out, int out_size, void* d_ws, size_t ws_size,
                              hipStream_t stream) {
  const float* x      = (const float*)d_in[0];
  const float* emb    = (const float*)d_in[1];
  const float* rms_w  = (const float*)d_in[2];
  const float* Win    = (const float*)d_in[3];
  const float* conv_w = (const float*)d_in[4];
  const float* conv_b = (const float*)d_in[5];
  const float* Wx     = (const float*)d_in[6];
  const float* Wdt    = (const float*)d_in[7];
  const float* bdt    = (const float*)d_in[8];
  const float* A_log  = (const float*)d_in[9];
  const float* Dp     = (const float*)d_in[10];
  const float* Wout   = (const float*)d_in[11];
  const float* Wq     = (const float*)d_in[12];
  const float* Wk     = (const float*)d_in[13];
  const float* Wv     = (const float*)d_in[14];
  const float* ln_s_g = (const float*)d_in[15];
  const float* ln_s_b = (const float*)d_in[16];
  const float* ln_c_g = (const float*)d_in[17];
  const float* ln_c_b = (const float*)d_in[18];
  const float* We     = (const float*)d_in[19];
  const float* be     = (const float*)d_in[20];
  const float* ln_h_g = (const float*)d_in[21];
  const float* ln_h_b = (const float*)d_in[22];
  const float* Wo     = (const float*)d_in[23];
  const float* bo     = (const float*)d_in[24];
  float* out = (float*)d_out;
  float* ws  = (float*)d_ws;

  // Workspace layout (floats); ~17.47M floats (~70 MB) with aliasing.
  float* XN    = ws;                 // 2048*768   (dead after GEMM1)
  float* XR    = ws + 1572864;       // 2048*3072
  float* UC    = ws + 7864320;       // 2048*1536
  float* XDBL  = ws + 11010048;      // 2048*80
  float* DELTA = ws + 11173888;      // 2048*1536  (dead after scan)
  float* YF    = ws + 14319616;      // 2048*1536
  float* EO    = ws + 17465344;      // 3072
  float* FOLD  = ws + 17468416;      // 864
  float* H     = XN;                 // reuse: 2048*768
  float* G     = DELTA;              // reuse: 2048*768

  rb_rms_kernel<<<MROWS, 256, 0, stream>>>(x, rms_w, XN);
  rb_gemm_xr_kernel<<<(128 * 192) / 8, 256, 0, stream>>>(XN, Win, XR);
  rb_conv_silu_kernel<<<(MROWS * DI) / 256, 256, 0, stream>>>(XR, conv_w, conv_b, UC);
  rb_gemm_xdbl_kernel<<<(128 * 5) / 8, 256, 0, stream>>>(UC, Wx, XDBL);
  rb_gemm_delta_kernel<<<(128 * 96) / 8, 256, 0, stream>>>(XDBL, Wdt, bdt, DELTA);
  rb_prep_fold_kernel<<<1, 64, 0, stream>>>(Wq, Wk, Wv, ln_s_g, ln_s_b, ln_c_g, ln_c_b, FOLD);
  rb_scan_kernel<<<12, 256, 0, stream>>>(DELTA, UC, XR, XDBL, A_log, Dp, FOLD, YF);
  rb_gemm_h_kernel<<<(128 * 48) / 8, 256, 0, stream>>>(YF, Wout, H);
  rb_emb_kernel<<<12, 256, 0, stream>>>(emb, We, be, EO);
  rb_lnfinal_kernel<<<MROWS, 256, 0, stream>>>(H, ln_h_g, ln_h_b, EO, G);
  rb_gemm_out_kernel<<<(128 * 48) / 8, 256, 0, stream>>>(G, Wo, x, bo, out);
}